// DurationBasedUpsample_21560735826325
// MI455X (gfx1250) — compile-verified
//
#include <hip/hip_runtime.h>

// Problem constants (fixed by the reference setup_inputs()).
#define B_   16
#define T_   512
#define D_   384
#define MAXLEN 4096

typedef float f4 __attribute__((ext_vector_type(4)));

// ---------------------------------------------------------------------------
// Kernel 1: per-batch duration rounding, inclusive prefix sum (LDS scan),
// lengths, and frame->source-row index table (searchsorted side='right').
// One block per batch, 512 threads (16 wave32s). Work is tiny (~8K elements
// total) — latency, not bandwidth, bound; LDS scan + CDNA5 split barriers.
// ---------------------------------------------------------------------------
__global__ __launch_bounds__(T_) void lr_scan_kernel(
    const float* __restrict__ dur,   // [B, T]
    int*         __restrict__ idx_ws,// [B, MAXLEN]  (-1 => masked/zero row)
    float*       __restrict__ len_out) // [B] lengths (stored as float)
{
  __shared__ int cum[T_];
  const int b = blockIdx.x;
  const int t = threadIdx.x;

  // n = max(1, round-half-even(dur))  — rintf is RNE, matching jnp.round.
  const float d = dur[b * T_ + t];
  int n = (int)rintf(d);
  n = n < 1 ? 1 : n;
  cum[t] = n;
  __syncthreads();

  // Hillis–Steele inclusive scan over 512 elements (9 steps).
  #pragma unroll
  for (int off = 1; off < T_; off <<= 1) {
    int v = cum[t];
    if (t >= off) v += cum[t - off];
    __syncthreads();
    cum[t] = v;
    __syncthreads();
  }

  const int length = cum[T_ - 1];
  if (t == 0) len_out[b] = (float)length;  // exactly representable (<= 4096)

  // Each thread resolves MAXLEN/T_ = 8 frames via binary search over cum.
  // searchsorted(cum, f, 'right') == count of cum[j] <= f == first j with cum[j] > f.
  #pragma unroll
  for (int k = 0; k < MAXLEN / T_; ++k) {
    const int f = t + k * T_;
    int lo = 0, hi = T_;
    while (lo < hi) {
      const int mid = (lo + hi) >> 1;
      if (cum[mid] <= f) lo = mid + 1; else hi = mid;
    }
    int idx = lo < (T_ - 1) ? lo : (T_ - 1);
    idx_ws[b * MAXLEN + f] = (f < length) ? idx : -1;  // fold mask into index
  }
}

// ---------------------------------------------------------------------------
// Kernel 2: streaming gather. One wave32 per output row, 2 rows per wave so
// we can prefetch row1's source while copying row0. Each row is 384 f32 =
// 96 float4 = 32 lanes x 3 x b128 — fully coalesced. Output is written with
// non-temporal hints (write-once 100 MB stream; keep enc_out hot in cache).
// Every row is written (data or zeros), so d_out is fully initialized.
// ---------------------------------------------------------------------------
__device__ __forceinline__ void lr_copy_row(const float* __restrict__ enc,
                                            float* __restrict__ out,
                                            int row, int idx, int lane)
{
  f4* __restrict__ orow = (f4*)(out + (size_t)row * D_);
  if (idx < 0) {
    const f4 z = {0.f, 0.f, 0.f, 0.f};
    #pragma unroll
    for (int k = 0; k < 3; ++k)
      __builtin_nontemporal_store(z, orow + lane + 32 * k);
  } else {
    const int b = row >> 12;  // MAXLEN = 4096 rows per batch
    const f4* __restrict__ erow =
        (const f4*)(enc + ((size_t)b * T_ + idx) * D_);
    #pragma unroll
    for (int k = 0; k < 3; ++k) {
      const f4 v = erow[lane + 32 * k];
      __builtin_nontemporal_store(v, orow + lane + 32 * k);
    }
  }
}

__global__ __launch_bounds__(256) void lr_gather_kernel(
    const float* __restrict__ enc,    // [B, T, D]
    const int*   __restrict__ idx_ws, // [B, MAXLEN]
    float*       __restrict__ out)    // [B, MAXLEN, D]
{
  const int lane = threadIdx.x & 31;
  const int wave = threadIdx.x >> 5;
  const int row0 = (blockIdx.x * 8 + wave) * 2;  // 8 waves/block, 2 rows/wave
  const int row1 = row0 + 1;

  const int i0 = idx_ws[row0];
  const int i1 = idx_ws[row1];

  // Prefetch row1's source line(s) into cache while we stream row0
  // (lowers to global_prefetch_b8 on gfx1250).
  if (i1 >= 0) {
    const int b1 = row1 >> 12;
    const float* src1 = enc + ((size_t)b1 * T_ + i1) * D_;
    __builtin_prefetch((const void*)(src1 + lane * 4), 0, 3);
  }

  lr_copy_row(enc, out, row0, i0, lane);
  lr_copy_row(enc, out, row1, i1, lane);
}

// ---------------------------------------------------------------------------
// Host-side launcher.
//   d_in[0] = enc_out  [16, 512, 384] f32
//   d_in[1] = durations[16, 512]      f32
//   d_out   = out [16, 4096, 384] f32  ++  lengths [16] (written as float)
//   d_ws    = int idx table [16 * 4096] = 256 KB
// ---------------------------------------------------------------------------
extern "C" void kernel_launch(void* const* d_in, const int* in_sizes, int n_in,
                              void* d_out, int out_size, void* d_ws, size_t ws_size,
                              hipStream_t stream) {
  (void)in_sizes; (void)n_in; (void)out_size; (void)ws_size;

  const float* enc = (const float*)d_in[0];
  const float* dur = (const float*)d_in[1];
  float* out       = (float*)d_out;
  int*   idx_ws    = (int*)d_ws;
  float* len_out   = out + (size_t)B_ * MAXLEN * D_;  // tuple output #2

  lr_scan_kernel<<<B_, T_, 0, stream>>>(dur, idx_ws, len_out);

  // B*MAXLEN = 65536 rows; 16 rows per 256-thread block -> 4096 blocks.
  lr_gather_kernel<<<(B_ * MAXLEN) / 16, 256, 0, stream>>>(enc, idx_ws, out);
}